// Router_13572096655869
// MI455X (gfx1250) — compile-verified
//
#include <hip/hip_runtime.h>
#include <math.h>
#include <stdint.h>

// ---------------------------------------------------------------------------
// MoE router for MI455X (gfx1250, wave32):
//   logits = x @ gate_w^T   (32768x4096 @ 4096x64, fp32) via v_wmma_f32_16x16x4_f32
//   staging via Tensor Data Mover (tensor_load_to_lds), double-buffered LDS
//   top-2 + renormalized softmax weights (softmax denominator cancels).
// ---------------------------------------------------------------------------

typedef float v2f __attribute__((ext_vector_type(2)));
typedef float v8f __attribute__((ext_vector_type(8)));
typedef unsigned int v4u __attribute__((ext_vector_type(4)));
typedef int v8i __attribute__((ext_vector_type(8)));
typedef int v4i __attribute__((ext_vector_type(4)));

#define TOKENS   32768        // 8 * 4096
#define DDIM     4096
#define NEXP     64
#define KC       128          // K chunk staged in LDS per stage
#define PAD      4            // bank-conflict padding (floats); TDM pads rows
#define LDW      (KC + PAD)   // 132 floats per staged row (132 % 64 == 4)
#define WAVES    4            // waves per workgroup
#define ROWS     16           // token rows per wave (one WMMA M tile)
#define LOGW     68           // padded stride for logits scratch (68 % 64 == 4)
#define NCHUNK   (DDIM / KC)  // 32
#define BUFSZ_G  (NEXP * LDW)          // floats per gate buffer
#define BUFSZ_X  (WAVES * ROWS * LDW)  // floats per x buffer

// Issue one TDM 2D tile load: 16 x 128 fp32 elements, row stride 4096 elems,
// padded in LDS to LDW-float rows (pad_interval=6 -> every 128 DWORDs,
// pad_amount=3 -> 4 DWORDs).
__device__ __forceinline__ void tdm_load_tile(uint32_t lds_off, const float* gptr)
{
    uint64_t ga   = (uint64_t)(uintptr_t)gptr;
    uint32_t galo = __builtin_amdgcn_readfirstlane((uint32_t)ga);
    uint32_t gahi = __builtin_amdgcn_readfirstlane((uint32_t)(ga >> 32));
    uint32_t loff = __builtin_amdgcn_readfirstlane(lds_off);

    v4u g0;
    g0.x = 1u;                                     // count=1 (valid user descriptor)
    g0.y = loff;                                   // lds_addr (bytes), bits 63:32
    g0.z = galo;                                   // global_addr[31:0],  bits 95:64
    g0.w = (gahi & 0x01FFFFFFu) | (2u << 30);      // global_addr[56:32] | type=2

    v8i g1;
    g1[0] = (int)((2u << 16)        // data_size = 2 -> 4 bytes
                | (1u << 20)        // pad_enable
                | (6u << 22)        // pad_interval: 2^(6+1)=128 DWORDs
                | (3u << 25));      // pad_amount: 4 DWORDs
    g1[1] = (int)(4096u << 16);     // tensor_dim0 = 4096 (low 16 bits at 63:48)
    g1[2] = (int)((uint32_t)ROWS << 16); // tensor_dim1 = 16 (low 16 bits at 95:80)
    g1[3] = (int)(128u << 16);      // tile_dim0 = 128 (bits 127:112)
    g1[4] = ROWS;                   // tile_dim1 = 16  (bits 143:128)
    g1[5] = 4096;                   // tensor_dim0_stride = 4096 (bits 191:160)
    g1[6] = 0;
    g1[7] = 0;

    v4i z4 = {0, 0, 0, 0};          // groups 2/3 unused (2D tensor)
    v8i z8 = {0, 0, 0, 0, 0, 0, 0, 0};
    __builtin_amdgcn_tensor_load_to_lds(g0, g1, z4, z4, z8, /*cpol=*/0);
}

__global__ __launch_bounds__(WAVES * 32, 1)
void router_wmma_tdm_kernel(const float* __restrict__ xp,
                            const float* __restrict__ gw,
                            int*   __restrict__ idx_out,
                            float* __restrict__ w_out,
                            float* __restrict__ logits_out)
{
    __shared__ float ldsG[2][BUFSZ_G];   // gate_w K-chunks: 2 x 64 x 132 (~66 KB)
    __shared__ float ldsX[2][BUFSZ_X];   // x K-chunks:     2 x 4 x 16 x 132 (~66 KB)

    const int tid  = threadIdx.x;
    const int lane = tid & 31;
    const int wave = tid >> 5;
    const int l    = lane & 15;       // row-within-tile / expert-within-ntile
    const int half = lane >> 4;       // selects K sub-pair of the fragment

    const int tileBase = (blockIdx.x * WAVES + wave) * ROWS;   // first token of tile

    // per-wave global bases
    const float* xg = xp + (size_t)tileBase * DDIM;        // this wave's 16 token rows
    const float* gg = gw + (size_t)(wave * 16) * DDIM;     // this wave's 16 expert rows

    // per-wave LDS byte offsets (low 32 bits of flat address == LDS offset)
    const uint32_t xOff0 = (uint32_t)(uintptr_t)&ldsX[0][wave * ROWS * LDW];
    const uint32_t gOff0 = (uint32_t)(uintptr_t)&ldsG[0][wave * 16 * LDW];
    const uint32_t xBufStride = (uint32_t)(BUFSZ_X * sizeof(float));
    const uint32_t gBufStride = (uint32_t)(BUFSZ_G * sizeof(float));

    v8f acc[4] = {};   // 4 N-tiles of 16 experts -> 16x64 fp32 logits per wave

    // ---- prologue: kick off chunk 0 into buffer 0 ----
    tdm_load_tile(xOff0, xg);
    tdm_load_tile(gOff0, gg);

    for (int c = 0; c < NCHUNK; ++c) {
        const int buf = c & 1;

        __builtin_amdgcn_s_wait_tensorcnt(0);   // my TDMs for chunk c are done
        __syncthreads();                        // everyone's TDMs are done

        // issue next chunk into the other buffer while we compute on this one
        if (c + 1 < NCHUNK) {
            const int nb = (c + 1) & 1;
            const int kcn = (c + 1) * KC;
            tdm_load_tile(xOff0 + nb * xBufStride, xg + kcn);
            tdm_load_tile(gOff0 + nb * gBufStride, gg + kcn);
        }

        // ---- GEMM inner loop: 32 K-steps of v_wmma_f32_16x16x4_f32 ----
        const float* Abase = &ldsX[buf][wave * ROWS * LDW] + l * LDW + 2 * half;
        const float* Bbase = &ldsG[buf][0] + l * LDW + 2 * half;
        #pragma unroll 4
        for (int k = 0; k < KC; k += 4) {
            v2f a = *(const v2f*)(Abase + k);
            #pragma unroll
            for (int n = 0; n < 4; ++n) {
                v2f b = *(const v2f*)(Bbase + n * 16 * LDW + k);
                acc[n] = __builtin_amdgcn_wmma_f32_16x16x4_f32(
                    /*neg_a=*/false, a, /*neg_b=*/false, b,
                    /*c_mod=*/(short)0, acc[n],
                    /*reuse_a=*/false, /*reuse_b=*/false);
            }
        }
        __syncthreads();   // all waves done reading buf before it is refilled
    }

    // ---- epilogue: spill logits (global + LDS transpose scratch) ----
    float* wlog = &ldsX[0][wave * ROWS * LDW];   // reuse buffer 0 region: 16 x LOGW
    #pragma unroll
    for (int n = 0; n < 4; ++n) {
        const int e = n * 16 + l;
        #pragma unroll
        for (int j = 0; j < 8; ++j) {
            const int m = j + 8 * half;                 // token row within tile
            const float v = acc[n][j];
            logits_out[(size_t)(tileBase + m) * NEXP + e] = v;
            wlog[m * LOGW + e] = v;
        }
    }
    __syncthreads();

    // ---- top-2 + renormalized softmax weights, one token per lane 0..15 ----
    if (lane < 16) {
        const int tok = tileBase + lane;
        const float* row = wlog + lane * LOGW;

        float m1 = -3.402823466e+38f, m2 = -3.402823466e+38f;
        int   i1 = 0, i2 = 0;
        #pragma unroll 8
        for (int e = 0; e < NEXP; ++e) {
            const float v = row[e];
            if (v > m1) { m2 = m1; i2 = i1; m1 = v; i1 = e; }
            else if (v > m2) { m2 = v; i2 = e; }
        }
        // softmax denominator cancels in the top-2 renormalization:
        // w1 = 1/(1+exp(m2-m1)), w2 = 1-w1
        const float t  = __expf(m2 - m1);
        const float w1 = 1.0f / (1.0f + t);
        const float w2 = 1.0f - w1;

        idx_out[tok * 2 + 0] = i1;
        idx_out[tok * 2 + 1] = i2;
        w_out  [tok * 2 + 0] = w1;
        w_out  [tok * 2 + 1] = w2;
    }
}

extern "C" void kernel_launch(void* const* d_in, const int* in_sizes, int n_in,
                              void* d_out, int out_size, void* d_ws, size_t ws_size,
                              hipStream_t stream)
{
    (void)in_sizes; (void)n_in; (void)out_size; (void)d_ws; (void)ws_size;

    const float* x  = (const float*)d_in[0];   // [8,4096,4096] f32
    const float* gw = (const float*)d_in[1];   // [64,4096] f32

    // d_out layout (tuple flattened in return order):
    //   [0 .. 65536)        indices (int32 bit-patterns)
    //   [65536 .. 131072)   weights (f32)
    //   [131072 .. end)     logits  (f32, 32768 x 64)
    float* out     = (float*)d_out;
    int*   idx_out = (int*)out;
    float* w_out   = out + (size_t)TOKENS * 2;
    float* lg_out  = out + (size_t)TOKENS * 4;

    const int tiles  = TOKENS / ROWS;          // 2048
    const int blocks = tiles / WAVES;          // 512
    router_wmma_tdm_kernel<<<dim3(blocks), dim3(WAVES * 32), 0, stream>>>(
        x, gw, idx_out, w_out, lg_out);
}